// GNN_nonstatic_13675175870742
// MI455X (gfx1250) — compile-verified
//
#include <hip/hip_runtime.h>

// Problem constants (from the reference)
#define BB      4
#define NN      20000
#define DD      128
#define HH      3
#define CCH     64
#define HC      192            // H*C
#define E0E     320000
#define BNN     (BB * NN)      // 80000
#define EBATCH  (BB * E0E)     // 1280000
#define ETOT    (EBATCH + BNN) // 1360000 (batched edges + self loops)
#define NEG_SLOPE 0.2f
#define NT      (HC / 16)      // 12 N-tiles per row stripe

typedef float v2f __attribute__((ext_vector_type(2)));
typedef float v8f __attribute__((ext_vector_type(8)));

// ---------------------------------------------------------------------------
// Order-preserving float <-> uint mapping for atomicMax-based segment max.
// encode maps all finite floats to (0, 0xFFFFFFFF] monotonically; a
// zero-initialized buffer therefore acts as -infinity.
// ---------------------------------------------------------------------------
__device__ __forceinline__ unsigned enc_f(float f) {
    unsigned u = __float_as_uint(f);
    return (u & 0x80000000u) ? ~u : (u | 0x80000000u);
}
__device__ __forceinline__ float dec_f(unsigned u) {
    return __uint_as_float((u & 0x80000000u) ? (u & 0x7fffffffu) : ~u);
}

// ---------------------------------------------------------------------------
// Decode edge id -> (src, dst) node ids (batched graph + self loops)
// ---------------------------------------------------------------------------
__device__ __forceinline__ void decode_edge(const int* __restrict__ ei, int e,
                                            int& src, int& dst) {
    if (e < EBATCH) {
        int b = e / E0E;
        int k = e - b * E0E;
        src = ei[k]       + b * NN;
        dst = ei[E0E + k] + b * NN;
    } else {
        src = dst = e - EBATCH;
    }
}

// ---------------------------------------------------------------------------
// Y[BN,192] = X[BN,128] @ W[128,192] + bias   via V_WMMA_F32_16X16X4_F32
//
// One wave computes a full 16x192 output stripe: the A fragment (16x4 of X)
// is loaded ONCE per k-step and reused across all 12 N-tiles (12x reduction
// in X traffic vs tile-per-block). 12 v8f accumulators = 96 VGPRs.
//
// A frag (16x4 f32): lane = {m = lane&15, koff = (lane>>4)*2}, holds K=koff,koff+1
// B frag (4x16 f32): lane = {n = lane&15, koff}, holds rows K=koff,koff+1
// C/D (16x16 f32):   acc[j][v] = Y[16*bx + v + 8*(lane>>4)][16*j + (lane&15)]
// ---------------------------------------------------------------------------
__global__ __launch_bounds__(32) void gemm_wmma_f32(
    const float* __restrict__ X, const float* __restrict__ W,
    const float* __restrict__ bias, float* __restrict__ Y)
{
    const int lane = threadIdx.x & 31;
    const int row0 = blockIdx.x * 16;
    const int m    = lane & 15;
    const int koff = (lane >> 4) * 2;

    v8f acc[NT];
#pragma unroll
    for (int j = 0; j < NT; ++j) acc[j] = (v8f){};

    const float* arow = X + (size_t)(row0 + m) * DD + koff;

#pragma unroll 2
    for (int k0 = 0; k0 < DD; k0 += 4) {
        v2f a = *(const v2f*)(arow + k0);          // 8B aligned (koff even)
        const float* w0 = W + (size_t)(k0 + koff)     * HC + m;
        const float* w1 = W + (size_t)(k0 + koff + 1) * HC + m;
#pragma unroll
        for (int j = 0; j < NT; ++j) {
            v2f b;
            b.x = w0[16 * j];
            b.y = w1[16 * j];
            acc[j] = __builtin_amdgcn_wmma_f32_16x16x4_f32(
                /*neg_a=*/false, a, /*neg_b=*/false, b,
                /*c_mod=*/(short)0, acc[j], /*reuse_a=*/false, /*reuse_b=*/false);
        }
    }

#pragma unroll
    for (int j = 0; j < NT; ++j) {
        const int col = 16 * j + m;
        const float bv = bias[col];
#pragma unroll
        for (int v = 0; v < 8; ++v) {
            int row = row0 + v + 8 * (lane >> 4);
            Y[(size_t)row * HC + col] = acc[j][v] + bv;
        }
    }
}

// ---------------------------------------------------------------------------
// out = bias (broadcast), denom = 0, segment-max buffer = 0 (== -inf encoded)
// ---------------------------------------------------------------------------
__global__ void init_out(float* __restrict__ out, const float* __restrict__ bias,
                         float* __restrict__ denom, unsigned* __restrict__ mbuf)
{
    int i = blockIdx.x * blockDim.x + threadIdx.x;
    if (i < BNN * HC) out[i] = bias[i % HC];
    if (i < BNN * HH) { denom[i] = 0.0f; mbuf[i] = 0u; }
}

// ---------------------------------------------------------------------------
// Per-edge attention logits + segment max.  One wave per edge; lane l handles
// channels l, l+32, ..., l+160 (coalesced 128B per step).  Channels 0-63 are
// head 0, 64-127 head 1, 128-191 head 2 -> each lane holds exactly 2 per head.
// ---------------------------------------------------------------------------
__global__ void edge_score(const int* __restrict__ ei,
                           const float* __restrict__ xl,
                           const float* __restrict__ xr,
                           const float* __restrict__ att,
                           float* __restrict__ score,
                           unsigned* __restrict__ mbuf)
{
    int wid  = blockIdx.x * (blockDim.x >> 5) + (threadIdx.x >> 5);
    int lane = threadIdx.x & 31;
    if (wid >= ETOT) return;

    int src, dst;
    decode_edge(ei, wid, src, dst);

    const float* pl = xl + (size_t)src * HC;
    const float* pr = xr + (size_t)dst * HC;

    float p0 = 0.f, p1 = 0.f, p2 = 0.f;
#pragma unroll
    for (int i = 0; i < 6; ++i) {
        int ch = lane + 32 * i;
        float v = pl[ch] + pr[ch];
        v = (v > 0.f) ? v : NEG_SLOPE * v;   // LeakyReLU(0.2)
        v *= att[ch];
        if (i < 2)      p0 += v;
        else if (i < 4) p1 += v;
        else            p2 += v;
    }
    // wave32 butterfly reduction
#pragma unroll
    for (int off = 16; off; off >>= 1) {
        p0 += __shfl_xor(p0, off, 32);
        p1 += __shfl_xor(p1, off, 32);
        p2 += __shfl_xor(p2, off, 32);
    }
    if (lane == 0) {
        size_t sb = (size_t)wid * 3;
        score[sb + 0] = p0; score[sb + 1] = p1; score[sb + 2] = p2;
        atomicMax(&mbuf[dst * 3 + 0], enc_f(p0));
        atomicMax(&mbuf[dst * 3 + 1], enc_f(p1));
        atomicMax(&mbuf[dst * 3 + 2], enc_f(p2));
    }
}

// ---------------------------------------------------------------------------
// ex = exp(score - max[dst]); denom[dst] += ex  (score overwritten with ex)
// ---------------------------------------------------------------------------
__global__ void edge_exp(const int* __restrict__ ei,
                         float* __restrict__ score,
                         const unsigned* __restrict__ mbuf,
                         float* __restrict__ denom)
{
    int e = blockIdx.x * blockDim.x + threadIdx.x;
    if (e >= ETOT) return;
    int src, dst;
    decode_edge(ei, e, src, dst);
    (void)src;
#pragma unroll
    for (int h = 0; h < 3; ++h) {
        float m  = dec_f(mbuf[dst * 3 + h]);
        float ex = __expf(score[(size_t)e * 3 + h] - m);
        score[(size_t)e * 3 + h] = ex;
        atomicAdd(&denom[dst * 3 + h], ex);
    }
}

// ---------------------------------------------------------------------------
// out[dst] += xl[src] * alpha    (wave per edge, float atomics, coalesced)
// ---------------------------------------------------------------------------
__global__ void edge_scatter(const int* __restrict__ ei,
                             const float* __restrict__ xl,
                             const float* __restrict__ score,
                             const float* __restrict__ denom,
                             float* __restrict__ out)
{
    int wid  = blockIdx.x * (blockDim.x >> 5) + (threadIdx.x >> 5);
    int lane = threadIdx.x & 31;
    if (wid >= ETOT) return;

    int src, dst;
    decode_edge(ei, wid, src, dst);

    size_t sb = (size_t)wid * 3;
    float a0 = score[sb + 0] / denom[dst * 3 + 0];
    float a1 = score[sb + 1] / denom[dst * 3 + 1];
    float a2 = score[sb + 2] / denom[dst * 3 + 2];

    const float* pl = xl + (size_t)src * HC;
    float*       po = out + (size_t)dst * HC;
#pragma unroll
    for (int i = 0; i < 6; ++i) {
        int ch = lane + 32 * i;
        float al = (i < 2) ? a0 : ((i < 4) ? a1 : a2);
        atomicAdd(&po[ch], pl[ch] * al);
    }
}

// ---------------------------------------------------------------------------
// Launch
// ---------------------------------------------------------------------------
extern "C" void kernel_launch(void* const* d_in, const int* in_sizes, int n_in,
                              void* d_out, int out_size, void* d_ws, size_t ws_size,
                              hipStream_t stream) {
    const float* x    = (const float*)d_in[0];   // [B,N,D]
    const int*   ei   = (const int*)  d_in[1];   // [2,E0] (int indices < N)
    const float* lw   = (const float*)d_in[2];   // [D, H*C]
    const float* lb   = (const float*)d_in[3];   // [H*C]
    const float* rw   = (const float*)d_in[4];   // [D, H*C]
    const float* rb   = (const float*)d_in[5];   // [H*C]
    const float* att  = (const float*)d_in[6];   // [H,C] flat = [192]
    const float* bias = (const float*)d_in[7];   // [H*C]
    float* out = (float*)d_out;                  // [B,N,H*C]

    // workspace carve-up (floats): xl | xr | score(ex) | denom | mbuf
    float*    xl    = (float*)d_ws;
    float*    xr    = xl    + (size_t)BNN * HC;
    float*    score = xr    + (size_t)BNN * HC;
    float*    denom = score + (size_t)ETOT * 3;
    unsigned* mbuf  = (unsigned*)(denom + (size_t)BNN * 3);

    // 1) node feature transforms via fp32 WMMA (exact fp32 accumulation);
    //    one wave per 16-row stripe, A fragment reused across all 12 N-tiles.
    gemm_wmma_f32<<<BNN / 16, 32, 0, stream>>>(x, lw, lb, xl);
    gemm_wmma_f32<<<BNN / 16, 32, 0, stream>>>(x, rw, rb, xr);

    // 2) init output(=bias), denom(=0), segment max(= -inf encoded as 0)
    init_out<<<(BNN * HC + 255) / 256, 256, 0, stream>>>(out, bias, denom, mbuf);

    // 3) edge logits + segment max (wave per edge)
    edge_score<<<(ETOT + 7) / 8, 256, 0, stream>>>(ei, xl, xr, att, score, mbuf);

    // 4) exp + segment sum (thread per edge)
    edge_exp<<<(ETOT + 255) / 256, 256, 0, stream>>>(ei, score, mbuf, denom);

    // 5) normalized message scatter (wave per edge)
    edge_scatter<<<(ETOT + 7) / 8, 256, 0, stream>>>(ei, xl, score, denom, out);
}